// HGTNodeClassifier_19413252177998
// MI455X (gfx1250) — compile-verified
//
#include <hip/hip_runtime.h>
#include <math.h>

#define NN 50000
#define EE 640000
#define DD 128
#define HH 4
#define DKK 32
#define RSQRT_DK 0.17677669529663687f
#define LN_EPS 1e-5f

typedef float v2f __attribute__((ext_vector_type(2)));
typedef float v8f __attribute__((ext_vector_type(8)));

// D = A(16x4 f32) * B(4x16 f32) + C(16x16 f32)  -- native fp32 WMMA on gfx1250
static __device__ __forceinline__ v8f wmma4(v2f a, v2f b, v8f c) {
  return __builtin_amdgcn_wmma_f32_16x16x4_f32(false, a, false, b, (short)0, c, false, false);
}

// ---------------- init workspace ----------------
__global__ __launch_bounds__(256) void k_init(float* mseg, float* denom, float* agg,
                                              unsigned* deg, float* gcsum, unsigned* counters) {
  int i = blockIdx.x * 256 + threadIdx.x;
  if (i < NN * HH) { mseg[i] = -INFINITY; denom[i] = 0.f; }
  if (i < NN * DD) agg[i] = 0.f;
  if (i < NN) deg[i] = 0u;
  if (i < DD) gcsum[i] = 0.f;
  if (i < 2) counters[i] = 0u;
}

// ---------------- per-row inverse norms (wave per row) ----------------
__global__ __launch_bounds__(256) void k_norm(const float* __restrict__ h, float* __restrict__ inv_norm) {
  int gwave = (blockIdx.x * 256 + threadIdx.x) >> 5;
  int lane = threadIdx.x & 31;
  if (gwave >= NN) return;
  const float* row = h + (size_t)gwave * DD;
  float s = 0.f;
#pragma unroll
  for (int c = 0; c < 4; c++) { float x = row[lane + 32 * c]; s += x * x; }
#pragma unroll
  for (int o = 16; o >= 1; o >>= 1) s += __shfl_xor(s, o);
  if (lane == 0) inv_norm[gwave] = 1.0f / fmaxf(sqrtf(s), 1e-12f);
}

// ---------------- degree scatter ----------------
__global__ __launch_bounds__(256) void k_deg(const int* __restrict__ src, const int* __restrict__ dst,
                                             unsigned* __restrict__ deg) {
  int i = blockIdx.x * 256 + threadIdx.x;
  if (i >= EE) return;
  atomicAdd(&deg[src[i]], 1u);
  atomicAdd(&deg[dst[i]], 1u);
}

// ---------------- count zero-degree nodes (wave ballot + one atomic) ----------------
__global__ __launch_bounds__(256) void k_zerocnt(const unsigned* __restrict__ deg, unsigned* __restrict__ counters) {
  int i = blockIdx.x * 256 + threadIdx.x;
  bool z = (i < NN) && (deg[i] == 0u);
  unsigned long long b = __ballot(z);
  if ((threadIdx.x & 31) == 0 && b) atomicAdd(&counters[0], (unsigned)__popcll(b));
}

// ---------------- fp32 WMMA GEMM: out[r,:] = A[r,:] @ W + bias, W is 128x128 ----------------
// block = 256 thr (8 waves). Block covers 64 rows x 128 cols. Wave -> (row strip, col half).
// Tail handling is block-uniform: only the last block takes the guarded store path.
__global__ __launch_bounds__(256) void k_gemm128(const float* __restrict__ A, const float* __restrict__ W,
                                                 const float* __restrict__ bias, float* __restrict__ out,
                                                 int nrows) {
  __shared__ float lds_w[DD * DD];  // 64 KB: weight tile resident in LDS
  int tid = threadIdx.x;
  for (int i = tid; i < DD * DD / 4; i += 256) ((float4*)lds_w)[i] = ((const float4*)W)[i];
  __syncthreads();
  int wave = tid >> 5, lane = tid & 31;
  int strip = wave >> 1, chalf = wave & 1;
  int m16 = lane & 15;
  int khi = (lane >> 4) << 1;  // K sub-offset: lanes 0-15 -> {0,1}, lanes 16-31 -> {2,3}
  int hi8 = (lane >> 4) << 3;  // C rows: +8 for upper lane half
  int arow_i = blockIdx.x * 64 + strip * 16 + m16;
  if (arow_i > nrows - 1) arow_i = nrows - 1;  // branch-free clamp (stores handle the tail)
  const float* arow = A + (size_t)arow_i * DD;
  v8f acc[4] = {};
  for (int kk = 0; kk < DD; kk += 4) {
    v2f a; a.x = arow[kk + khi]; a.y = arow[kk + khi + 1];
#pragma unroll
    for (int t = 0; t < 4; t++) {
      int n = chalf * 64 + t * 16 + m16;
      v2f b; b.x = lds_w[(kk + khi) * DD + n]; b.y = lds_w[(kk + khi + 1) * DD + n];
      acc[t] = wmma4(a, b, acc[t]);
    }
  }
  int rbase = blockIdx.x * 64 + strip * 16 + hi8;  // rows rbase..rbase+7 for this lane half
  if (blockIdx.x * 64 + 64 <= nrows) {
    // full tile: branch-free coalesced stores (base + immediate row offsets)
#pragma unroll
    for (int t = 0; t < 4; t++) {
      int n = chalf * 64 + t * 16 + m16;
      float bn = bias[n];
      float* o = out + (size_t)rbase * DD + n;
#pragma unroll
      for (int j = 0; j < 8; j++) o[(size_t)j * DD] = acc[t][j] + bn;
    }
  } else {
#pragma unroll
    for (int t = 0; t < 4; t++) {
      int n = chalf * 64 + t * 16 + m16;
      float bn = bias[n];
#pragma unroll
      for (int j = 0; j < 8; j++) {
        int r = rbase + j;
        if (r < nrows) out[(size_t)r * DD + n] = acc[t][j] + bn;
      }
    }
  }
}

// ---------------- edge attention logits: WMMA relation transforms + q.k dot + boost ----------------
// wave handles 16 edges; per head: [16x32] @ rel[32x32] for both q and k (16 wmma each).
__global__ __launch_bounds__(256) void k_edge_att(
    const float* __restrict__ kbuf, const float* __restrict__ qbuf,
    const float* __restrict__ h, const float* __restrict__ inv_norm,
    const int* __restrict__ src, const int* __restrict__ dst,
    const float* __restrict__ rel_k, const float* __restrict__ rel_q,
    const unsigned* __restrict__ counters, const float* __restrict__ fsb,
    float* __restrict__ att, float* __restrict__ mseg) {
  __shared__ float lds_rk[HH * DKK * DKK];  // 16 KB
  __shared__ float lds_rq[HH * DKK * DKK];  // 16 KB
  __shared__ float lds_fs[8 * 16];          // feat_sim per wave-tile
  int tid = threadIdx.x;
  for (int i = tid; i < HH * DKK * DKK; i += 256) { lds_rk[i] = rel_k[i]; lds_rq[i] = rel_q[i]; }
  __syncthreads();
  int wave = tid >> 5, lane = tid & 31;
  int m16 = lane & 15, khi = (lane >> 4) << 1, hi8 = (lane >> 4) << 3;
  int e0 = (blockIdx.x * 8 + wave) * 16;
  float boost = ((float)counters[0] * (1.0f / NN) > 0.3f) ? fsb[0] : 0.0f;
  int eL = e0 + m16;
  int se = src[eL], de = dst[eL];
  // feature-similarity boost term: full wave cooperates per edge (dims split 4-per-lane)
  for (int j = 0; j < 16; j++) {
    int sj = __shfl(se, j);
    int dj = __shfl(de, j);
    const float* hs = h + (size_t)sj * DD;
    const float* hd = h + (size_t)dj * DD;
    float acc = 0.f;
#pragma unroll
    for (int c = 0; c < 4; c++) acc += hs[lane + 32 * c] * hd[lane + 32 * c];
#pragma unroll
    for (int o = 16; o >= 1; o >>= 1) acc += __shfl_xor(acc, o);
    if (lane == 0) lds_fs[wave * 16 + j] = acc * inv_norm[sj] * inv_norm[dj];
  }
  for (int hd = 0; hd < HH; hd++) {
    const float* rq = lds_rq + hd * DKK * DKK;
    const float* rk = lds_rk + hd * DKK * DKK;
    const float* qrow = qbuf + (size_t)de * DD + hd * DKK;
    const float* krow = kbuf + (size_t)se * DD + hd * DKK;
    v8f qa0 = {}, qa1 = {}, ka0 = {}, ka1 = {};
    for (int kk = 0; kk < DKK; kk += 4) {
      v2f aq; aq.x = qrow[kk + khi]; aq.y = qrow[kk + khi + 1];
      v2f ak; ak.x = krow[kk + khi]; ak.y = krow[kk + khi + 1];
      v2f b;
      b.x = rq[(kk + khi) * DKK + m16];      b.y = rq[(kk + khi + 1) * DKK + m16];
      qa0 = wmma4(aq, b, qa0);
      b.x = rq[(kk + khi) * DKK + 16 + m16]; b.y = rq[(kk + khi + 1) * DKK + 16 + m16];
      qa1 = wmma4(aq, b, qa1);
      b.x = rk[(kk + khi) * DKK + m16];      b.y = rk[(kk + khi + 1) * DKK + m16];
      ka0 = wmma4(ak, b, ka0);
      b.x = rk[(kk + khi) * DKK + 16 + m16]; b.y = rk[(kk + khi + 1) * DKK + 16 + m16];
      ka1 = wmma4(ak, b, ka1);
    }
#pragma unroll
    for (int j = 0; j < 8; j++) {
      float p = qa0[j] * ka0[j] + qa1[j] * ka1[j];           // row-wise q_t . k_t
      p += __shfl_xor(p, 8); p += __shfl_xor(p, 4);
      p += __shfl_xor(p, 2); p += __shfl_xor(p, 1);          // reduce within 16-lane half
      if (m16 == 0) {
        int em = e0 + j + hi8;
        float aval = p * RSQRT_DK + boost * lds_fs[wave * 16 + j + hi8];
        att[(size_t)em * HH + hd] = aval;
        int dm = dst[em];
        __hip_atomic_fetch_max(&mseg[dm * HH + hd], aval, __ATOMIC_RELAXED, __HIP_MEMORY_SCOPE_AGENT);
      }
    }
  }
}

// ---------------- exp(att - m[dst]) and segment denominator ----------------
__global__ __launch_bounds__(256) void k_edge_exp(const int* __restrict__ dst,
                                                  float* __restrict__ att, const float* __restrict__ mseg,
                                                  float* __restrict__ denom) {
  int i = blockIdx.x * 256 + threadIdx.x;  // over E*H
  if (i >= EE * HH) return;
  int e = i >> 2, hd = i & 3;
  int d = dst[e];
  float x = __expf(att[i] - mseg[d * HH + hd]);
  att[i] = x;
  atomicAdd(&denom[d * HH + hd], x);
}

// ---------------- value relation transform (WMMA) + weighted scatter-add ----------------
__global__ __launch_bounds__(256) void k_edge_agg(
    const float* __restrict__ vbuf, const int* __restrict__ src, const int* __restrict__ dst,
    const float* __restrict__ rel_v, const float* __restrict__ att,
    const float* __restrict__ denom, float* __restrict__ agg) {
  __shared__ float lds_rv[HH * DKK * DKK];
  int tid = threadIdx.x;
  for (int i = tid; i < HH * DKK * DKK; i += 256) lds_rv[i] = rel_v[i];
  __syncthreads();
  int wave = tid >> 5, lane = tid & 31;
  int m16 = lane & 15, khi = (lane >> 4) << 1, hi8 = (lane >> 4) << 3;
  int e0 = (blockIdx.x * 8 + wave) * 16;
  int se = src[e0 + m16];
  for (int hd = 0; hd < HH; hd++) {
    const float* rv = lds_rv + hd * DKK * DKK;
    const float* vrow = vbuf + (size_t)se * DD + hd * DKK;
    v8f a0 = {}, a1 = {};
    for (int kk = 0; kk < DKK; kk += 4) {
      v2f av; av.x = vrow[kk + khi]; av.y = vrow[kk + khi + 1];
      v2f b;
      b.x = rv[(kk + khi) * DKK + m16];      b.y = rv[(kk + khi + 1) * DKK + m16];
      a0 = wmma4(av, b, a0);
      b.x = rv[(kk + khi) * DKK + 16 + m16]; b.y = rv[(kk + khi + 1) * DKK + 16 + m16];
      a1 = wmma4(av, b, a1);
    }
#pragma unroll
    for (int j = 0; j < 8; j++) {
      int em = e0 + j + hi8;
      int dm = dst[em];
      float w = att[(size_t)em * HH + hd] / denom[dm * HH + hd];
      float* arow = agg + (size_t)dm * DD + hd * DKK;
      atomicAdd(&arow[m16], a0[j] * w);
      atomicAdd(&arow[16 + m16], a1[j] * w);
    }
  }
}

// ---------------- LayerNorm in-place + block-local gc reduction ----------------
__global__ __launch_bounds__(256) void k_ln(float* __restrict__ out,
                                            const float* __restrict__ ln_g, const float* __restrict__ ln_b,
                                            const unsigned* __restrict__ deg, float* __restrict__ gcsum,
                                            unsigned* __restrict__ counters) {
  __shared__ float lsum[DD];
  __shared__ unsigned lcnt;
  int tid = threadIdx.x;
  if (tid < DD) lsum[tid] = 0.f;
  if (tid == 0) lcnt = 0u;
  __syncthreads();
  int wave = tid >> 5, lane = tid & 31;
  for (int it = 0; it < 8; it++) {
    int row = blockIdx.x * 64 + it * 8 + wave;
    if (row >= NN) continue;
    float* r = out + (size_t)row * DD;
    float x[4]; float s = 0.f;
#pragma unroll
    for (int c = 0; c < 4; c++) { x[c] = r[lane + 32 * c]; s += x[c]; }
#pragma unroll
    for (int o = 16; o >= 1; o >>= 1) s += __shfl_xor(s, o);
    float mu = s * (1.0f / DD);
    float vs = 0.f;
#pragma unroll
    for (int c = 0; c < 4; c++) { float d2 = x[c] - mu; vs += d2 * d2; }
#pragma unroll
    for (int o = 16; o >= 1; o >>= 1) vs += __shfl_xor(vs, o);
    float inv = rsqrtf(vs * (1.0f / DD) + LN_EPS);
    bool nz = deg[row] != 0u;
#pragma unroll
    for (int c = 0; c < 4; c++) {
      int col = lane + 32 * c;
      float y = (x[c] - mu) * inv * ln_g[col] + ln_b[col];
      r[col] = y;
      if (nz) atomicAdd(&lsum[col], y);  // LDS ds_add_f32
    }
    if (nz && lane == 0) atomicAdd(&lcnt, 1u);
  }
  __syncthreads();
  if (tid < DD) atomicAdd(&gcsum[tid], lsum[tid]);
  if (tid == 0) atomicAdd(&counters[1], lcnt);
}

// ---------------- zero-degree fixup: 0.9*h + 0.1*gc ----------------
__global__ __launch_bounds__(256) void k_fixup(const float* __restrict__ h,
                                               const unsigned* __restrict__ deg, const float* __restrict__ gcsum,
                                               const unsigned* __restrict__ counters, float* __restrict__ out) {
  int i = blockIdx.x * 256 + threadIdx.x;
  if (i >= NN * DD) return;
  int row = i >> 7, col = i & 127;
  if (deg[row] == 0u) {
    float cnt = fmaxf((float)counters[1], 1.0f);
    out[i] = 0.9f * h[i] + 0.1f * (gcsum[col] / cnt);
  }
}

extern "C" void kernel_launch(void* const* d_in, const int* in_sizes, int n_in,
                              void* d_out, int out_size, void* d_ws, size_t ws_size,
                              hipStream_t stream) {
  const float* h = (const float*)d_in[0];
  const int* src = (const int*)d_in[1];
  const int* dst = (const int*)d_in[2];
  const float* Wk = (const float*)d_in[3];
  const float* bk = (const float*)d_in[4];
  const float* Wq = (const float*)d_in[5];
  const float* bq = (const float*)d_in[6];
  const float* Wv = (const float*)d_in[7];
  const float* bv = (const float*)d_in[8];
  const float* rel_k = (const float*)d_in[9];
  const float* rel_q = (const float*)d_in[10];
  const float* rel_v = (const float*)d_in[11];
  const float* Wa = (const float*)d_in[12];
  const float* ba = (const float*)d_in[13];
  const float* ln_g = (const float*)d_in[14];
  const float* ln_b = (const float*)d_in[15];
  const float* fsb = (const float*)d_in[16];
  float* out = (float*)d_out;

  float* ws = (float*)d_ws;
  float* kbuf = ws;      ws += (size_t)NN * DD;
  float* qbuf = ws;      ws += (size_t)NN * DD;
  float* vbuf = ws;      ws += (size_t)NN * DD;
  float* agg = ws;       ws += (size_t)NN * DD;
  float* att = ws;       ws += (size_t)EE * HH;
  float* mseg = ws;      ws += (size_t)NN * HH;
  float* denom = ws;     ws += (size_t)NN * HH;
  float* inv_norm = ws;  ws += NN;
  float* gcsum = ws;     ws += DD;
  unsigned* deg = (unsigned*)ws;       ws += NN;
  unsigned* counters = (unsigned*)ws;  // [0]=zero-degree count, [1]=nz-row count

  dim3 b256(256);
  k_init<<<(NN * DD + 255) / 256, b256, 0, stream>>>(mseg, denom, agg, deg, gcsum, counters);
  k_norm<<<(NN * 32 + 255) / 256, b256, 0, stream>>>(h, inv_norm);
  k_deg<<<(EE + 255) / 256, b256, 0, stream>>>(src, dst, deg);
  k_zerocnt<<<(NN + 255) / 256, b256, 0, stream>>>(deg, counters);
  k_gemm128<<<(NN + 63) / 64, b256, 0, stream>>>(h, Wk, bk, kbuf, NN);
  k_gemm128<<<(NN + 63) / 64, b256, 0, stream>>>(h, Wq, bq, qbuf, NN);
  k_gemm128<<<(NN + 63) / 64, b256, 0, stream>>>(h, Wv, bv, vbuf, NN);
  k_edge_att<<<EE / 128, b256, 0, stream>>>(kbuf, qbuf, h, inv_norm, src, dst, rel_k, rel_q,
                                            counters, fsb, att, mseg);
  k_edge_exp<<<(EE * HH + 255) / 256, b256, 0, stream>>>(dst, att, mseg, denom);
  k_edge_agg<<<EE / 128, b256, 0, stream>>>(vbuf, src, dst, rel_v, att, denom, agg);
  k_gemm128<<<(NN + 63) / 64, b256, 0, stream>>>(agg, Wa, ba, out, NN);
  k_ln<<<(NN + 63) / 64, b256, 0, stream>>>(out, ln_g, ln_b, deg, gcsum, counters);
  k_fixup<<<(NN * DD + 255) / 256, b256, 0, stream>>>(h, deg, gcsum, counters, out);
}